// LoRABasisBank_79302276153534
// MI455X (gfx1250) — compile-verified
//
#include <hip/hip_runtime.h>

typedef __attribute__((ext_vector_type(2))) float v2f;
typedef __attribute__((ext_vector_type(8))) float v8f;

#if defined(__HIP_DEVICE_COMPILE__)
#if !__has_builtin(__builtin_amdgcn_wmma_f32_16x16x4_f32)
#error "device pass: missing __builtin_amdgcn_wmma_f32_16x16x4_f32"
#endif
#endif

// D = A(16x4 f32) x B(4x16 f32) + C(16x16 f32), wave32 WMMA.
__device__ __forceinline__ v8f wmma_f32_16x16x4(v2f a, v2f b, v8f c) {
#if defined(__HIP_DEVICE_COMPILE__)
  return __builtin_amdgcn_wmma_f32_16x16x4_f32(false, a, false, b,
                                               (short)0, c, false, false);
#else
  (void)a; (void)b;
  return c; // host pass: never executed, just needs to parse
#endif
}

namespace {
constexpr int Bn = 8, Sn = 2048, Kn = 16, Rn = 16, DIN = 2048, DOUT = 2048;
constexpr int WAVES = 4;                   // waves per block (wave32)
constexpr int ROWS_PER_BLOCK = WAVES * 16; // s-rows per workgroup
}

// Fold alpha into the banks, storing WMMA-B-fragment-friendly layouts:
//   APair[b][i/2][r][2]  (pair of consecutive i per r)  -> stage-1 B frags
//   BPair[b][r/2][o][2]  (pair of consecutive r per o)  -> stage-2 B frags
__global__ __launch_bounds__(256) void lora_mix(
    const float* __restrict__ alpha,   // [B][K]
    const float* __restrict__ A_bank,  // [K][R][DIN]
    const float* __restrict__ B_bank,  // [K][DOUT][R]
    float* __restrict__ APair,
    float* __restrict__ BPair)
{
  const int NA = Bn * Rn * DIN;
  int idx = blockIdx.x * blockDim.x + threadIdx.x;
  if (idx < NA) {
    int i = idx % DIN;
    int r = (idx / DIN) % Rn;
    int b = idx / (DIN * Rn);
    float acc = 0.f;
#pragma unroll
    for (int k = 0; k < Kn; ++k)
      acc = fmaf(alpha[b * Kn + k], A_bank[((size_t)k * Rn + r) * DIN + i], acc);
    APair[(((size_t)b * (DIN / 2) + (i >> 1)) * Rn + r) * 2 + (i & 1)] = acc;
  } else {
    int j = idx - NA;
    int o = j % DOUT;
    int r = (j / DOUT) % Rn;
    int b = j / (DOUT * Rn);
    if (b >= Bn) return;
    float acc = 0.f;
#pragma unroll
    for (int k = 0; k < Kn; ++k)
      acc = fmaf(alpha[b * Kn + k], B_bank[((size_t)k * DOUT + o) * Rn + r], acc);
    BPair[(((size_t)b * (Rn / 2) + (r >> 1)) * DOUT + o) * 2 + (r & 1)] = acc;
  }
}

// One wave owns a 16-row s-tile:
//   stage 1: z(16x16)  = h_tile(16xDIN)   x A_mixed^T(DINx16)   [512 wmma]
//   stage 2: d(16x16k) = z(16x16)         x B_mixed^T(16xDOUT)  [4 wmma / 16 cols]
__global__ __launch_bounds__(WAVES * 32) void lora_apply(
    const float* __restrict__ h,      // [B][S][DIN]
    const float* __restrict__ APair,  // [B][DIN/2][R][2]
    const float* __restrict__ BPair,  // [B][R/2][DOUT][2]
    float* __restrict__ delta)        // [B][S][DOUT]
{
  __shared__ float zsh[WAVES][16][17]; // +1 pad: conflict-free column gather

  const int lane = threadIdx.x & 31;
  const int wave = threadIdx.x >> 5;
  const int half = lane >> 4;   // lane 0-15 vs 16-31
  const int l16  = lane & 15;
  const int b    = blockIdx.y;
  const int row0 = blockIdx.x * ROWS_PER_BLOCK + wave * 16;

  // ---- stage 1: accumulate z over DIN ----
  const float* hrow = h + ((size_t)b * Sn + row0 + l16) * DIN;       // A frag rows
  const v2f*   ap   = (const v2f*)(APair + (size_t)b * DIN * Rn);    // [i2*Rn + r]

  v8f z = {};
#pragma unroll 4
  for (int k0 = 0; k0 < DIN; k0 += 4) {
    // A frag (16x4 f32): vgpr j, lane -> h[row=l16][k0 + 2*half + j]  (one b64)
    v2f a = *(const v2f*)(hrow + k0 + 2 * half);
    // B frag (4x16 f32): vgpr j, lane -> A_mixed[r=l16][i=k0 + 2*half + j] (one b64)
    v2f bf = ap[(size_t)(k0 / 2 + half) * Rn + l16];
    z = wmma_f32_16x16x4(a, bf, z);
  }

  // ---- relayout z: C-layout -> A-fragment layout via LDS ----
#pragma unroll
  for (int j = 0; j < 8; ++j)
    zsh[wave][j + 8 * half][l16] = z[j];   // row = M, col = N
  __syncthreads();

  v2f za[4]; // stage-2 A frags, K chunks of 4 over r, reused for every o-tile
#pragma unroll
  for (int c = 0; c < 4; ++c) {
    za[c].x = zsh[wave][l16][4 * c + 2 * half + 0];
    za[c].y = zsh[wave][l16][4 * c + 2 * half + 1];
  }

  // ---- stage 2: sweep DOUT in 16-wide tiles ----
  const v2f* bp = (const v2f*)(BPair + (size_t)b * Rn * DOUT);       // [r2*DOUT + o]
  float* drow = delta + ((size_t)b * Sn + row0) * DOUT;

  for (int o0 = 0; o0 < DOUT; o0 += 16) {
    v8f d = {};
#pragma unroll
    for (int c = 0; c < 4; ++c) {
      // B frag: vgpr j, lane -> B_mixed[o=o0+l16][r=4c + 2*half + j]  (one b64)
      v2f bf = bp[(size_t)(2 * c + half) * DOUT + o0 + l16];
      d = wmma_f32_16x16x4(za[c], bf, d);
    }
#pragma unroll
    for (int j = 0; j < 8; ++j) // D layout: M = j + 8*half, N = l16 (coalesced b32)
      drow[(size_t)(j + 8 * half) * DOUT + o0 + l16] = d[j];
  }
}

extern "C" void kernel_launch(void* const* d_in, const int* in_sizes, int n_in,
                              void* d_out, int out_size, void* d_ws, size_t ws_size,
                              hipStream_t stream) {
  const float* h      = (const float*)d_in[0];
  const float* alpha  = (const float*)d_in[1];
  const float* A_bank = (const float*)d_in[2];
  const float* B_bank = (const float*)d_in[3];
  float* delta = (float*)d_out;

  float* APair = (float*)d_ws;                         // 1 MB
  float* BPair = APair + (size_t)Bn * DIN * Rn;        // 1 MB

  const int total = Bn * Rn * DIN + Bn * Rn * DOUT;    // mix-kernel threads
  lora_mix<<<(total + 255) / 256, 256, 0, stream>>>(alpha, A_bank, B_bank,
                                                    APair, BPair);

  dim3 grid(Sn / ROWS_PER_BLOCK, Bn);
  lora_apply<<<grid, WAVES * 32, 0, stream>>>(h, APair, BPair, delta);
}